// PlaneModel_16209206575263
// MI455X (gfx1250) — compile-verified
//
#include <hip/hip_runtime.h>
#include <hip/hip_bf16.h>
#include <math.h>

// ---------------- problem constants (from reference) ----------------
#define BATCH   8192
#define NV      8
#define NVOX    512          // NV^3
#define P_FEAT  64
#define NS      32
#define BASE    32
#define HID1    16           // BASE/2
#define OUT_N   13
#define D_IN    79           // 3 + P + 12
#define D_PAD   80           // padded K (multiple of 4)
#define LN_EPS  1e-6f

// ---------------- kernel tiling ----------------
#define WAVES_PER_BLOCK 4
#define BLOCK_THREADS   (WAVES_PER_BLOCK * 32)
#define ZSTRIDE         84   // 80 cols + 4 pad floats; conflict-free 16-row strides, 16B-aligned rows

typedef float v2f  __attribute__((ext_vector_type(2)));
typedef float v8f  __attribute__((ext_vector_type(8)));
typedef int   v4i  __attribute__((ext_vector_type(4)));

// CDNA5 async direct-to-LDS gather path (gfx1250): memory -> LDS with no VGPR staging.
#if defined(__has_builtin)
#  if __has_builtin(__builtin_amdgcn_global_load_async_to_lds_b128)
#    define HAVE_ASYNC_LDS 1
#  endif
#endif
#ifndef HAVE_ASYNC_LDS
#  define HAVE_ASYNC_LDS 0
#endif

#if HAVE_ASYNC_LDS
typedef __attribute__((address_space(1))) v4i gv4i;   // global int4
typedef __attribute__((address_space(3))) v4i lv4i;   // LDS int4
#endif

__global__ __launch_bounds__(BLOCK_THREADS)
void plane_model_fused(const float* __restrict__ pos,
                       const float* __restrict__ quat,
                       const float* __restrict__ xz,
                       const float* __restrict__ aabb,
                       const int*   __restrict__ widx,
                       const float* __restrict__ pos_log_scale,
                       const float* __restrict__ w1,
                       const float* __restrict__ b1,
                       const float* __restrict__ g1,
                       const float* __restrict__ be1,
                       const float* __restrict__ w2a,
                       const float* __restrict__ b2a,
                       const float* __restrict__ g2a,
                       const float* __restrict__ be2a,
                       const float* __restrict__ w2b,
                       const float* __restrict__ b2b,
                       const float* __restrict__ g2b,
                       const float* __restrict__ be2b,
                       const float* __restrict__ wout,
                       const float* __restrict__ bout,
                       float* __restrict__ out)
{
    // LDS: permuted w1 (80x16) + per-wave z staging (32 x ZSTRIDE)
    __shared__ __attribute__((aligned(16))) float w1s[D_PAD * HID1];
    __shared__ __attribute__((aligned(16))) float zbuf[WAVES_PER_BLOCK][NS * ZSTRIDE];

    const int tid  = threadIdx.x;
    const int wave = tid >> 5;
    const int lane = tid & 31;
    const int b    = blockIdx.x * WAVES_PER_BLOCK + wave;

    // ---- per-lane: kick off the async 256B xz-row gather straight into LDS
    //      z cols: [0..63]=xz, [64..66]=pick, [67..78]=pose, [79]=0-pad
    const int s = lane;
    const int w = widx[b * NS + s];                 // 0..511
    float* row = &zbuf[wave][s * ZSTRIDE];
    {
        const char* gsrc = (const char*)(xz + ((size_t)b * NVOX + (size_t)w) * P_FEAT);
#if HAVE_ASYNC_LDS
        char* ldst = (char*)row;
        #pragma unroll
        for (int t = 0; t < 16; ++t) {
            __builtin_amdgcn_global_load_async_to_lds_b128(
                (gv4i*)(gsrc + t * 16), (lv4i*)(ldst + t * 16),
                /*offset=*/0, /*cpol=*/0);
        }
#else
        const float4* src4 = (const float4*)gsrc;
        float4* dst4 = (float4*)row;
        #pragma unroll
        for (int t = 0; t < 16; ++t) dst4[t] = src4[t];
#endif
    }

    // ---- stage w1 into LDS with rows permuted to match our z column order
    //      w1 src rows (reference order): [0..2]=pick, [3..66]=xz, [67..78]=pose
    for (int idx = tid; idx < D_PAD * HID1; idx += BLOCK_THREADS) {
        const int r  = idx >> 4;
        const int nn = idx & 15;
        int src;
        if      (r < 64) src = r + 3;     // xz rows
        else if (r < 67) src = r - 64;    // pick rows
        else if (r < 79) src = r;         // pose rows
        else             src = -1;        // pad row
        w1s[idx] = (src >= 0) ? w1[src * HID1 + nn] : 0.0f;
    }

    // ---- per-lane: pick / pose columns (overlaps with async gather in flight)
    {
        const int ci = (w >> 6) & 7, cj = (w >> 3) & 7, ck = w & 7;

        // quaternion -> R (row-major 9)
        float q0 = quat[b*4+0], q1 = quat[b*4+1], q2 = quat[b*4+2], q3 = quat[b*4+3];
        const float qinv = rsqrtf(q0*q0 + q1*q1 + q2*q2 + q3*q3);
        const float x = q0*qinv, y = q1*qinv, z = q2*qinv, wq = q3*qinv;
        float R0 = 1.f - 2.f*(y*y + z*z);
        float R1 = 2.f*(x*y - z*wq);
        float R2 = 2.f*(x*z + y*wq);
        float R3 = 2.f*(x*y + z*wq);
        float R4 = 1.f - 2.f*(x*x + z*z);
        float R5 = 2.f*(y*z - x*wq);
        float R6 = 2.f*(x*z - y*wq);
        float R7 = 2.f*(y*z + x*wq);
        float R8 = 1.f - 2.f*(x*x + y*y);

        const float p0 = pos[b*3+0], p1 = pos[b*3+1], p2 = pos[b*3+2];
        const float lo0 = aabb[b*6+0], lo1 = aabb[b*6+1], lo2 = aabb[b*6+2];
        const float hi0 = aabb[b*6+3], hi1 = aabb[b*6+4], hi2 = aabb[b*6+5];

        const float v0 = ((float)ci + 0.5f) * 0.125f * (hi0 - lo0) + lo0;
        const float v1 = ((float)cj + 0.5f) * 0.125f * (hi1 - lo1) + lo1;
        const float v2 = ((float)ck + 0.5f) * 0.125f * (hi2 - lo2) + lo2;

        const float pk0 = R0*v0 + R1*v1 + R2*v2 + p0;
        const float pk1 = R3*v0 + R4*v1 + R5*v2 + p1;
        const float pk2 = R6*v0 + R7*v1 + R8*v2 + p2;

        const float es = __expf(pos_log_scale[0]);

        row[64] = pk0; row[65] = pk1; row[66] = pk2;
        row[67] = p0 * es; row[68] = p1 * es; row[69] = p2 * es;
        row[70] = R0; row[71] = R1; row[72] = R2;
        row[73] = R3; row[74] = R4; row[75] = R5;
        row[76] = R6; row[77] = R7; row[78] = R8;
        row[79] = 0.0f;
    }

#if HAVE_ASYNC_LDS
    // Drain this wave's async memory->LDS transfers before the block barrier.
    asm volatile("s_wait_asynccnt 0" ::: "memory");
#endif
    __syncthreads();

    // ---- layer 1 GEMM: z(32 x 80) @ w1(80 x 16) via V_WMMA_F32_16X16X4_F32
    // two 16-row M tiles, 20 K-steps of 4
    v8f c0 = {};
    v8f c1 = {};
    {
        const float* zw = &zbuf[wave][0];
        const int rA   = lane & 15;          // A-matrix row within tile
        const int kOff = (lane >> 4) * 2;    // lanes 16-31 hold K+2,K+3
        const int n    = lane & 15;          // B-matrix column

        #pragma unroll
        for (int k0 = 0; k0 < D_PAD; k0 += 4) {
            v2f a0 = *(const v2f*)(zw + rA * ZSTRIDE + k0 + kOff);
            v2f a1 = *(const v2f*)(zw + (rA + 16) * ZSTRIDE + k0 + kOff);
            v2f bf;
            bf.x = w1s[(k0 + kOff)     * HID1 + n];
            bf.y = w1s[(k0 + kOff + 1) * HID1 + n];
            c0 = __builtin_amdgcn_wmma_f32_16x16x4_f32(false, a0, false, bf,
                                                       (short)0, c0, false, false);
            c1 = __builtin_amdgcn_wmma_f32_16x16x4_f32(false, a1, false, bf,
                                                       (short)0, c1, false, false);
        }
    }

    // ---- relu + bias + LayerNorm(16) per row, x2, sum over the 32 rows
    // D layout: VGPR r = row r (lanes 0-15) / row r+8 (lanes 16-31); col = lane&15
    const int n16 = lane & 15;
    const float b1v  = b1[n16];
    const float g1v  = g1[n16];
    const float be1v = be1[n16];

    float acc = 0.0f;
    #pragma unroll
    for (int r = 0; r < 8; ++r) {
        #pragma unroll
        for (int t = 0; t < 2; ++t) {
            float xv = (t == 0) ? c0[r] : c1[r];
            xv = fmaxf(xv + b1v, 0.0f);
            // mean over the 16-lane half-wave
            float ssum = xv;
            ssum += __shfl_xor(ssum, 1);  ssum += __shfl_xor(ssum, 2);
            ssum += __shfl_xor(ssum, 4);  ssum += __shfl_xor(ssum, 8);
            const float mean = ssum * (1.0f / 16.0f);
            const float d = xv - mean;
            float vs = d * d;
            vs += __shfl_xor(vs, 1); vs += __shfl_xor(vs, 2);
            vs += __shfl_xor(vs, 4); vs += __shfl_xor(vs, 8);
            const float var = vs * (1.0f / 16.0f);
            const float yv = d * rsqrtf(var + LN_EPS) * g1v + be1v;
            acc += 2.0f * yv;   // (z + z) then sum over NS
        }
    }
    // combine the two half-wave row groups -> h[n] replicated in both halves
    const float h = acc + __shfl_xor(acc, 16);

    // ---- layer 2a: h(16) @ w2a(16x32) + b2a, relu, LN(32)
    const int cCol = lane;
    float s2 = b2a[cCol];
    #pragma unroll
    for (int k = 0; k < HID1; ++k)
        s2 = fmaf(__shfl(h, k), w2a[k * BASE + cCol], s2);
    s2 = fmaxf(s2, 0.0f);
    {
        float ssum = s2;
        ssum += __shfl_xor(ssum, 1);  ssum += __shfl_xor(ssum, 2);  ssum += __shfl_xor(ssum, 4);
        ssum += __shfl_xor(ssum, 8);  ssum += __shfl_xor(ssum, 16);
        const float mean = ssum * (1.0f / 32.0f);
        const float d = s2 - mean;
        float vs = d * d;
        vs += __shfl_xor(vs, 1); vs += __shfl_xor(vs, 2); vs += __shfl_xor(vs, 4);
        vs += __shfl_xor(vs, 8); vs += __shfl_xor(vs, 16);
        s2 = d * rsqrtf(vs * (1.0f / 32.0f) + LN_EPS) * g2a[cCol] + be2a[cCol];
    }
    const float skip = s2;

    // ---- layer 2b: z(32) @ w2b(32x32) + b2b, relu, LN(32), + skip
    float s3 = b2b[cCol];
    #pragma unroll
    for (int k = 0; k < BASE; ++k)
        s3 = fmaf(__shfl(s2, k), w2b[k * BASE + cCol], s3);
    s3 = fmaxf(s3, 0.0f);
    {
        float ssum = s3;
        ssum += __shfl_xor(ssum, 1);  ssum += __shfl_xor(ssum, 2);  ssum += __shfl_xor(ssum, 4);
        ssum += __shfl_xor(ssum, 8);  ssum += __shfl_xor(ssum, 16);
        const float mean = ssum * (1.0f / 32.0f);
        const float d = s3 - mean;
        float vs = d * d;
        vs += __shfl_xor(vs, 1); vs += __shfl_xor(vs, 2); vs += __shfl_xor(vs, 4);
        vs += __shfl_xor(vs, 8); vs += __shfl_xor(vs, 16);
        s3 = d * rsqrtf(vs * (1.0f / 32.0f) + LN_EPS) * g2b[cCol] + be2b[cCol];
    }
    const float zf = s3 + skip;

    // ---- output head: zf(32) @ wout(32x13) + bout, tanh
    const int o = (lane < OUT_N) ? lane : 0;
    float so = bout[o];
    #pragma unroll
    for (int k = 0; k < BASE; ++k)
        so = fmaf(__shfl(zf, k), wout[k * OUT_N + o], so);
    if (lane < OUT_N)
        out[(size_t)b * OUT_N + lane] = tanhf(so);
}

extern "C" void kernel_launch(void* const* d_in, const int* in_sizes, int n_in,
                              void* d_out, int out_size, void* d_ws, size_t ws_size,
                              hipStream_t stream) {
    (void)in_sizes; (void)n_in; (void)out_size; (void)d_ws; (void)ws_size;
    const float* pos  = (const float*)d_in[0];
    const float* quat = (const float*)d_in[1];
    const float* xz   = (const float*)d_in[2];
    const float* aabb = (const float*)d_in[3];
    const int*   widx = (const int*)  d_in[4];
    const float* pls  = (const float*)d_in[5];
    const float* w1   = (const float*)d_in[6];
    const float* b1   = (const float*)d_in[7];
    const float* g1   = (const float*)d_in[8];
    const float* be1  = (const float*)d_in[9];
    const float* w2a  = (const float*)d_in[10];
    const float* b2a  = (const float*)d_in[11];
    const float* g2a  = (const float*)d_in[12];
    const float* be2a = (const float*)d_in[13];
    const float* w2b  = (const float*)d_in[14];
    const float* b2b  = (const float*)d_in[15];
    const float* g2b  = (const float*)d_in[16];
    const float* be2b = (const float*)d_in[17];
    const float* wout = (const float*)d_in[18];
    const float* bout = (const float*)d_in[19];

    const int blocks = BATCH / WAVES_PER_BLOCK;   // 2048
    plane_model_fused<<<blocks, BLOCK_THREADS, 0, stream>>>(
        pos, quat, xz, aabb, widx, pls,
        w1, b1, g1, be1,
        w2a, b2a, g2a, be2a,
        w2b, b2b, g2b, be2b,
        wout, bout, (float*)d_out);
}